// SSMBlock_42331197669518
// MI455X (gfx1250) — compile-verified
//
#include <hip/hip_runtime.h>
#include <hip/hip_bf16.h>

typedef float v2f __attribute__((ext_vector_type(2)));
typedef float v8f __attribute__((ext_vector_type(8)));

#define DIM    1024
#define NSTATE 16
#define DTRANK 64
#define BATCH  2
#define SEQ    2048
#define ROWS   (BATCH * SEQ)   // 4096 token rows
#define NCHUNK 64              // scan chunks
#define CL     (SEQ / NCHUNK)  // 32 steps per chunk

__device__ __forceinline__ v2f ld2(const float* p) { return *(const v2f*)p; }

// ---------------------------------------------------------------------------
// fp32 WMMA GEMM:  C[M,N] = A[M,K] * W[N,K]^T   (row-major)
// One wave computes a 16x64 tile: one A fragment reused across 4 N-subtiles,
// 8 V_WMMA_F32_16X16X4_F32 per K-step-8 as two groups of 4 independent chains.
// EPIL==1: C = softplus(acc + bias[col])
// ---------------------------------------------------------------------------
template <int EPIL>
__global__ __launch_bounds__(128) void wmma_gemm_f32(
    const float* __restrict__ A, int lda,
    const float* __restrict__ W, int ldb,
    float* __restrict__ C, int ldc,
    int M, int N, int K,
    const float* __restrict__ bias)
{
    const int lane  = threadIdx.x & 31;
    const int wave  = threadIdx.x >> 5;
    const int m0    = blockIdx.x * 16;
    const int nbase = (blockIdx.y * 4 + wave) * 64;
    if (nbase >= N) return;

    const int mi = lane & 15;
    const int hi = lane >> 4;

    const float* arow = A + (size_t)(m0 + mi) * lda + 2 * hi;
    const float* brow[4];
    bool act[4];
#pragma unroll
    for (int j = 0; j < 4; ++j) {
        const int nj = nbase + 16 * j;
        act[j]  = (nj < N);
        brow[j] = W + (size_t)((act[j] ? nj : 0) + mi) * ldb + 2 * hi;
    }

    v8f acc0 = {}, acc1 = {}, acc2 = {}, acc3 = {};

    for (int kk = 0; kk < K; kk += 8) {
        __builtin_prefetch(arow + kk + 64, 0, 3);        // WGP-scope prefetch
        __builtin_prefetch(brow[0] + kk + 64, 0, 3);
        __builtin_prefetch(brow[1] + kk + 64, 0, 3);
        __builtin_prefetch(brow[2] + kk + 64, 0, 3);
        __builtin_prefetch(brow[3] + kk + 64, 0, 3);

        v2f a0 = ld2(arow + kk);
        v2f a1 = ld2(arow + kk + 4);
        v2f b00 = ld2(brow[0] + kk), b01 = ld2(brow[0] + kk + 4);
        v2f b10 = ld2(brow[1] + kk), b11 = ld2(brow[1] + kk + 4);
        v2f b20 = ld2(brow[2] + kk), b21 = ld2(brow[2] + kk + 4);
        v2f b30 = ld2(brow[3] + kk), b31 = ld2(brow[3] + kk + 4);

        acc0 = __builtin_amdgcn_wmma_f32_16x16x4_f32(false, a0, false, b00, (short)0, acc0, false, false);
        acc1 = __builtin_amdgcn_wmma_f32_16x16x4_f32(false, a0, false, b10, (short)0, acc1, false, false);
        acc2 = __builtin_amdgcn_wmma_f32_16x16x4_f32(false, a0, false, b20, (short)0, acc2, false, false);
        acc3 = __builtin_amdgcn_wmma_f32_16x16x4_f32(false, a0, false, b30, (short)0, acc3, false, false);
        acc0 = __builtin_amdgcn_wmma_f32_16x16x4_f32(false, a1, false, b01, (short)0, acc0, false, false);
        acc1 = __builtin_amdgcn_wmma_f32_16x16x4_f32(false, a1, false, b11, (short)0, acc1, false, false);
        acc2 = __builtin_amdgcn_wmma_f32_16x16x4_f32(false, a1, false, b21, (short)0, acc2, false, false);
        acc3 = __builtin_amdgcn_wmma_f32_16x16x4_f32(false, a1, false, b31, (short)0, acc3, false, false);
    }

    v8f accs[4] = {acc0, acc1, acc2, acc3};
#pragma unroll
    for (int j = 0; j < 4; ++j) {
        if (!act[j]) continue;
        const int col = nbase + 16 * j + mi;
#pragma unroll
        for (int r = 0; r < 8; ++r) {
            const int row = m0 + r + 8 * hi;
            float v = accs[j][r];
            if (EPIL == 1) {
                v += bias[col];
                v = (v > 20.f) ? v : log1pf(__expf(v));   // softplus
            }
            C[(size_t)row * ldc + col] = v;
        }
    }
}

// ---------------------------------------------------------------------------
// Depthwise conv1d (k=3, pad 1) + bias, reading the xc half of xz.
// ---------------------------------------------------------------------------
__global__ __launch_bounds__(256) void conv_kernel(
    const float* __restrict__ xz, const float* __restrict__ w,
    const float* __restrict__ b, float* __restrict__ xc)
{
    int idx = blockIdx.x * 256 + threadIdx.x;       // ROWS*DIM threads
    int d   = idx & (DIM - 1);
    int row = idx >> 10;
    int l   = row & (SEQ - 1);

    const float* wd  = w + d * 3;
    const float* col = xz + d;                      // stride 2*DIM per row

    float acc = b[d] + wd[1] * col[(size_t)row * (2 * DIM)];
    if (l > 0)       acc += wd[0] * col[(size_t)(row - 1) * (2 * DIM)];
    if (l < SEQ - 1) acc += wd[2] * col[(size_t)(row + 1) * (2 * DIM)];
    xc[idx] = acc;
}

// ---------------------------------------------------------------------------
// Chunked selective scan, pass A: per-(b,d,chunk) compute P[n] = prod(dA)
// and S[n] = chunk-local state (from zero init). Layout [b][chunk][n][d].
// ---------------------------------------------------------------------------
__global__ __launch_bounds__(256) void scan_partial(
    const float* __restrict__ xc, const float* __restrict__ xdbl,
    const float* __restrict__ delta, const float* __restrict__ A_log,
    float* __restrict__ P, float* __restrict__ S)
{
    __shared__ float sbc[32];
    const int t     = blockIdx.x * 256 + threadIdx.x;  // (b*NCHUNK+chunk)*DIM + d
    const int d     = t & (DIM - 1);
    const int chunk = (t >> 10) & (NCHUNK - 1);
    const int b     = t >> 16;

    float Arow[NSTATE];
#pragma unroll
    for (int n = 0; n < NSTATE; ++n) Arow[n] = -__expf(A_log[d * NSTATE + n]);

    float p[NSTATE], s[NSTATE];
#pragma unroll
    for (int n = 0; n < NSTATE; ++n) { p[n] = 1.f; s[n] = 0.f; }

    const int l0 = chunk * CL;
    for (int l = 0; l < CL; ++l) {
        const int row = b * SEQ + l0 + l;
        if (threadIdx.x < 32)
            sbc[threadIdx.x] = xdbl[(size_t)row * 96 + DTRANK + threadIdx.x];
        __syncthreads();
        const float dt = delta[(size_t)row * DIM + d];
        const float u  = xc[(size_t)row * DIM + d];
#pragma unroll
        for (int n = 0; n < NSTATE; ++n) {
            const float dA = __expf(dt * Arow[n]);
            s[n] = dA * s[n] + (dt * sbc[n]) * u;
            p[n] *= dA;
        }
        __syncthreads();
    }

    const size_t base = ((size_t)(b * NCHUNK + chunk) * NSTATE) * DIM + d;
#pragma unroll
    for (int n = 0; n < NSTATE; ++n) {
        P[base + (size_t)n * DIM] = p[n];
        S[base + (size_t)n * DIM] = s[n];
    }
}

// ---------------------------------------------------------------------------
// Pass B: sequential carry composition across the 64 chunks per (b,d,n).
// init[b][chunk][n][d] = state entering that chunk.
// ---------------------------------------------------------------------------
__global__ __launch_bounds__(256) void scan_prefix(
    const float* __restrict__ P, const float* __restrict__ S,
    float* __restrict__ init)
{
    const int t = blockIdx.x * 256 + threadIdx.x;   // 0..BATCH*DIM-1
    const int d = t & (DIM - 1);
    const int b = t >> 10;

    float carry[NSTATE];
#pragma unroll
    for (int n = 0; n < NSTATE; ++n) carry[n] = 0.f;

    for (int c = 0; c < NCHUNK; ++c) {
        const size_t base = ((size_t)(b * NCHUNK + c) * NSTATE) * DIM + d;
#pragma unroll
        for (int n = 0; n < NSTATE; ++n) {
            const size_t a = base + (size_t)n * DIM;
            init[a] = carry[n];
            carry[n] = P[a] * carry[n] + S[a];
        }
    }
}

// ---------------------------------------------------------------------------
// Pass C: replay each chunk from its correct initial state; emit
// y = C·state + D*u, gated by silu(z); written in place over xc.
// ---------------------------------------------------------------------------
__global__ __launch_bounds__(256) void scan_final(
    const float* __restrict__ xz, float* __restrict__ xc,
    const float* __restrict__ xdbl, const float* __restrict__ delta,
    const float* __restrict__ A_log, const float* __restrict__ D_param,
    const float* __restrict__ init)
{
    __shared__ float sbc[32];
    const int t     = blockIdx.x * 256 + threadIdx.x;
    const int d     = t & (DIM - 1);
    const int chunk = (t >> 10) & (NCHUNK - 1);
    const int b     = t >> 16;

    float Arow[NSTATE];
#pragma unroll
    for (int n = 0; n < NSTATE; ++n) Arow[n] = -__expf(A_log[d * NSTATE + n]);
    const float Dd = D_param[d];

    const size_t base = ((size_t)(b * NCHUNK + chunk) * NSTATE) * DIM + d;
    float st[NSTATE];
#pragma unroll
    for (int n = 0; n < NSTATE; ++n) st[n] = init[base + (size_t)n * DIM];

    const int l0 = chunk * CL;
    for (int l = 0; l < CL; ++l) {
        const int row = b * SEQ + l0 + l;
        if (threadIdx.x < 32)
            sbc[threadIdx.x] = xdbl[(size_t)row * 96 + DTRANK + threadIdx.x];
        __syncthreads();
        const float dt = delta[(size_t)row * DIM + d];
        const float u  = xc[(size_t)row * DIM + d];

        float y = 0.f;
#pragma unroll
        for (int n = 0; n < NSTATE; ++n) {
            const float dA = __expf(dt * Arow[n]);
            const float s  = dA * st[n] + (dt * sbc[n]) * u;
            st[n] = s;
            y += s * sbc[16 + n];
        }
        y += Dd * u;

        const float z   = xz[(size_t)row * (2 * DIM) + DIM + d];
        const float sil = z / (1.f + __expf(-z));
        __syncthreads();
        xc[(size_t)row * DIM + d] = y * sil;
    }
}

// ---------------------------------------------------------------------------
extern "C" void kernel_launch(void* const* d_in, const int* in_sizes, int n_in,
                              void* d_out, int out_size, void* d_ws, size_t ws_size,
                              hipStream_t stream) {
    const float* x          = (const float*)d_in[0];
    const float* in_proj_w  = (const float*)d_in[1];
    const float* conv_w     = (const float*)d_in[2];
    const float* conv_b     = (const float*)d_in[3];
    const float* A_log      = (const float*)d_in[4];
    const float* D_param    = (const float*)d_in[5];
    const float* x_proj_w   = (const float*)d_in[6];
    const float* dt_proj_w  = (const float*)d_in[7];
    const float* dt_proj_b  = (const float*)d_in[8];
    const float* out_proj_w = (const float*)d_in[9];
    float* out = (float*)d_out;

    // Workspace (floats): xz | xc | x_dbl | delta | P | S | init  ~= 89.5 MB
    float* xz   = (float*)d_ws;
    float* xc   = xz   + (size_t)ROWS * 2 * DIM;
    float* xdbl = xc   + (size_t)ROWS * DIM;
    float* delt = xdbl + (size_t)ROWS * 96;
    float* Pb   = delt + (size_t)ROWS * DIM;
    float* Sb   = Pb   + (size_t)BATCH * NCHUNK * NSTATE * DIM;
    float* initb= Sb   + (size_t)BATCH * NCHUNK * NSTATE * DIM;

    const dim3 blk(128);

    // 1) xz = x @ in_proj_w^T            (M=4096, N=2048, K=1024)
    wmma_gemm_f32<0><<<dim3(ROWS / 16, (2 * DIM) / 256), blk, 0, stream>>>(
        x, DIM, in_proj_w, DIM, xz, 2 * DIM, ROWS, 2 * DIM, DIM, nullptr);

    // 2) depthwise conv + bias -> xc
    conv_kernel<<<(ROWS * DIM) / 256, 256, 0, stream>>>(xz, conv_w, conv_b, xc);

    // 3) x_dbl = xc @ x_proj_w^T         (N=96)
    wmma_gemm_f32<0><<<dim3(ROWS / 16, 1), blk, 0, stream>>>(
        xc, DIM, x_proj_w, DIM, xdbl, 96, ROWS, 96, DIM, nullptr);

    // 4) delta = softplus(dt_r @ dt_proj_w^T + dt_proj_b)   (K=64)
    wmma_gemm_f32<1><<<dim3(ROWS / 16, DIM / 256), blk, 0, stream>>>(
        xdbl, 96, dt_proj_w, DTRANK, delt, DIM, ROWS, DIM, DTRANK, dt_proj_b);

    // 5) chunked selective scan (3 passes) + D*u + silu(z) gate, in place
    scan_partial<<<(BATCH * NCHUNK * DIM) / 256, 256, 0, stream>>>(
        xc, xdbl, delt, A_log, Pb, Sb);
    scan_prefix<<<(BATCH * DIM) / 256, 256, 0, stream>>>(Pb, Sb, initb);
    scan_final<<<(BATCH * NCHUNK * DIM) / 256, 256, 0, stream>>>(
        xz, xc, xdbl, delt, A_log, D_param, initb);

    // 6) out = yz @ out_proj_w^T
    wmma_gemm_f32<0><<<dim3(ROWS / 16, DIM / 256), blk, 0, stream>>>(
        xc, DIM, out_proj_w, DIM, out, DIM, ROWS, DIM, DIM, nullptr);
}